// OnsetLSTM_4389456577294
// MI455X (gfx1250) — compile-verified
//
#include <hip/hip_runtime.h>
#include <hip/hip_bf16.h>

// ---------------------------------------------------------------------------
// Fused 2-layer LSTM for gfx1250 (MI455X), fp32 WMMA (V_WMMA_F32_16X16X4_F32).
//
// One workgroup owns a 16-row batch tile and runs all 512 timesteps of both
// layers with state held in LDS/registers. Weights (~1.1MB) are packed once
// into WMMA-fragment order in d_ws and streamed from L2 every step.
//
// - `lz` laundering defeats LICM so weight loads stay in-loop (round 1/2).
// - acc starts at inline-0 and bias is added at gate-store time, so no
//   splatted-bias v8f registers stay live across the loop (round 3).
// - gemm_part double-buffers A (LDS) and B (global) fragments one ktp ahead
//   so L2 latency overlaps the WMMA chain instead of serializing on
//   s_wait_loadcnt 0x0 (round 3).
// ---------------------------------------------------------------------------

typedef float v2f __attribute__((ext_vector_type(2)));
typedef float v4f __attribute__((ext_vector_type(4)));
typedef float v8f __attribute__((ext_vector_type(8)));

// Problem constants (from reference)
constexpr int B  = 256;
constexpr int T  = 512;
constexpr int D  = 160;   // layer0 input features
constexpr int H  = 128;   // hidden
constexpr int GS = 516;   // padded gate row stride in LDS (bank-conflict free)

// K-tile-pair counts (each "ktp" covers K=8 = two K=4 WMMA fragments)
constexpr int KTP_X = D / 8;  // 20
constexpr int KTP_H = H / 8;  // 16

// Packed-weight workspace layout (float units)
constexpr int OFF_IH0 = 0;                       // 32nt*20ktp*32lane*4 = 81920
constexpr int OFF_HH0 = 81920;                   // 65536
constexpr int OFF_IH1 = 147456;                  // 65536
constexpr int OFF_HH1 = 212992;                  // 65536
constexpr int OFF_B0  = 278528;                  // 512
constexpr int OFF_B1  = 279040;                  // 512

// quad (v4f) offsets
constexpr int QIH0 = OFF_IH0 / 4;   // 0
constexpr int QHH0 = OFF_HH0 / 4;   // 20480
constexpr int QIH1 = OFF_IH1 / 4;   // 36864
constexpr int QHH1 = OFF_HH1 / 4;   // 53248
constexpr int QTOTAL = OFF_B0 / 4;  // 69632 weight quads

// ---------------------------------------------------------------------------
// helpers
// ---------------------------------------------------------------------------
__device__ __forceinline__ float fsig(float x) {
    // sigmoid(x) = 1/(1+exp(-x)); native v_exp_f32 (2^x) + v_rcp_f32
    float e = __builtin_amdgcn_exp2f(-1.442695041f * x);
    return __builtin_amdgcn_rcpf(1.0f + e);
}

__device__ __forceinline__ float ftanh(float x) {
    x = fminf(15.0f, fmaxf(-15.0f, x));
    float e = __builtin_amdgcn_exp2f(-2.885390082f * x);   // exp(-2x)
    return (1.0f - e) * __builtin_amdgcn_rcpf(1.0f + e);
}

__device__ __forceinline__ v8f wmma4(v2f a, v2f b, v8f c) {
    // V_WMMA_F32_16X16X4_F32 : D = A(16x4) * B(4x16) + C
    return __builtin_amdgcn_wmma_f32_16x16x4_f32(
        /*neg_a=*/false, a, /*neg_b=*/false, b,
        /*c_mod=*/(short)0, c, /*reuse_a=*/false, /*reuse_b=*/false);
}

__device__ __forceinline__ v2f vlo(v4f v) {
    return __builtin_shufflevector(v, v, 0, 1);
}
__device__ __forceinline__ v2f vhi(v4f v) {
    return __builtin_shufflevector(v, v, 2, 3);
}

// GEMM over one input operand: acc[nt] += Afrag(16 x 8*KTP) * Bfrags.
// A fragments in LDS as v4f[ktp*32 + lane] (pair of K=4 fragments per 16B);
// B fragments in global as v4f[(nt*KTP + ktp)*32 + lane].
// Double-buffered: next ktp's A/B fragments are fetched while the current
// ktp's 8 WMMAs run (full unroll renames the buffer swap away).
template <int KTP>
__device__ __forceinline__ void gemm_part(const v4f* __restrict__ aq,
                                          const v4f* __restrict__ pB,
                                          int lane, v8f (&acc)[4]) {
    v4f a = aq[lane];
    v4f b[4];
#pragma unroll
    for (int nt = 0; nt < 4; ++nt) b[nt] = pB[(nt * KTP) * 32 + lane];

#pragma unroll
    for (int ktp = 0; ktp < KTP; ++ktp) {
        v4f a_n = a;
        v4f b_n[4] = {b[0], b[1], b[2], b[3]};
        if (ktp + 1 < KTP) {
            a_n = aq[((ktp + 1) << 5) + lane];
#pragma unroll
            for (int nt = 0; nt < 4; ++nt)
                b_n[nt] = pB[(nt * KTP + ktp + 1) * 32 + lane];
        }
        v2f a0 = vlo(a), a1 = vhi(a);
#pragma unroll
        for (int nt = 0; nt < 4; ++nt) {
            acc[nt] = wmma4(a0, vlo(b[nt]), acc[nt]);
            acc[nt] = wmma4(a1, vhi(b[nt]), acc[nt]);
        }
        a = a_n;
#pragma unroll
        for (int nt = 0; nt < 4; ++nt) b[nt] = b_n[nt];
    }
}

// Scatter C/D accumulators (16x16 f32 layout) into the padded gate buffer,
// adding the per-column bias (scalar regs, not splatted v8f).
__device__ __forceinline__ void store_gates(const v8f (&acc)[4],
                                            const float (&biasr)[4],
                                            float* __restrict__ g,
                                            int w, int lane) {
    int col = (w << 6) + (lane & 15);
    int rb  = (lane >> 4) << 3;      // lanes 16-31 hold rows M=8..15
#pragma unroll
    for (int nt = 0; nt < 4; ++nt) {
#pragma unroll
        for (int v = 0; v < 8; ++v) {
            g[(rb + v) * GS + col + (nt << 4)] = acc[nt][v] + biasr[nt];
        }
    }
}

// LSTM cell elementwise update. Thread owns column jown, rows mb..mb+7.
// Reads raw gates from g (padded rows), keeps c in creg, writes h into the
// A-fragment-layout LDS buffer hfrag; optionally writes h to global out.
__device__ __forceinline__ void cell_update(const float* __restrict__ g,
                                            float* __restrict__ creg,
                                            float* __restrict__ hfrag,
                                            int jown, int mb,
                                            int ktpj, int lanej, int slotj,
                                            float* __restrict__ outp) {
#pragma unroll
    for (int i = 0; i < 8; ++i) {
        int m = mb + i;
        const float* gr = g + m * GS + jown;
        float gi = fsig(gr[0]);
        float gf = fsig(gr[128]);
        float gg = ftanh(gr[256]);
        float go = fsig(gr[384]);
        float c  = gf * creg[i] + gi * gg;
        creg[i]  = c;
        float h  = go * ftanh(c);
        hfrag[ktpj * 128 + (lanej + m) * 4 + slotj] = h;
        if (outp) outp[m * H + jown] = h;
    }
}

// ---------------------------------------------------------------------------
// One-time weight pack: B-fragment layout [nt][ktp][lane]{4 floats}
// lane: (lane&15)=n within tile, (lane>>4) selects k sub-pair.
// quad = { W[n][k0], W[n][k0+1], W[n][k0+4], W[n][k0+5] }, k0 = ktp*8 + (lane>>4)*2
// Also precomputes bias sums b_ih + b_hh.
// ---------------------------------------------------------------------------
__global__ __launch_bounds__(256) void lstm_pack(
    const float* __restrict__ Wih0, const float* __restrict__ Whh0,
    const float* __restrict__ Wih1, const float* __restrict__ Whh1,
    const float* __restrict__ bih0, const float* __restrict__ bhh0,
    const float* __restrict__ bih1, const float* __restrict__ bhh1,
    float* __restrict__ ws) {
    int id = blockIdx.x * 256 + threadIdx.x;
    if (id < QTOTAL) {
        const float* W;
        int K, KTP, local, qbase;
        if (id < QHH0)      { W = Wih0; K = D; KTP = KTP_X; local = id;         qbase = QIH0; }
        else if (id < QIH1) { W = Whh0; K = H; KTP = KTP_H; local = id - QHH0;  qbase = QHH0; }
        else if (id < QHH1) { W = Wih1; K = H; KTP = KTP_H; local = id - QIH1;  qbase = QIH1; }
        else                { W = Whh1; K = H; KTP = KTP_H; local = id - QHH1;  qbase = QHH1; }
        int per_nt = KTP * 32;
        int nt   = local / per_nt;
        int r    = local - nt * per_nt;
        int ktp  = r >> 5;
        int lane = r & 31;
        int n    = nt * 16 + (lane & 15);
        int k0   = ktp * 8 + ((lane >> 4) << 1);
        const float* row = W + n * K + k0;
        v4f v = {row[0], row[1], row[4], row[5]};
        ((v4f*)ws)[qbase + local] = v;
    } else if (id < QTOTAL + 1024) {
        int j = id - QTOTAL;
        if (j < 512) ws[OFF_B0 + j]         = bih0[j] + bhh0[j];
        else         ws[OFF_B1 + (j - 512)] = bih1[j - 512] + bhh1[j - 512];
    }
}

// ---------------------------------------------------------------------------
// Fused persistent 2-layer LSTM. Grid: 16 blocks x 256 threads (8 wave32).
// Each block owns batch rows [b0, b0+16). Wave w computes gate columns
// [64w, 64w+64) (4 N-tiles) via f32 WMMA.
// ---------------------------------------------------------------------------
__global__ __launch_bounds__(256) void lstm_fused(
    const float* __restrict__ x,     // [B, T, D]
    const float* __restrict__ h0in,  // [2, B, H]
    const float* __restrict__ c0in,  // [2, B, H]
    const float* __restrict__ ws,    // packed weights + biases
    float* __restrict__ out) {       // [B, 1, H]
    __shared__ float sm_g[16 * GS];          // raw gates   (33.0 KB)
    __shared__ v4f   sm_x [KTP_X * 32];      // x_t  frags  (10.0 KB)
    __shared__ v4f   sm_h0[KTP_H * 32];      // h(l0) frags ( 8.0 KB)
    __shared__ v4f   sm_h1[KTP_H * 32];      // h(l1) frags ( 8.0 KB)

    const int tid  = threadIdx.x;
    const int lane = tid & 31;
    const int w    = tid >> 5;
    const int b0   = blockIdx.x << 4;

    const v4f* wsq   = (const v4f*)ws;
    const v4f* pBih0 = wsq + QIH0 + w * (4 * KTP_X * 32);
    const v4f* pBhh0 = wsq + QHH0 + w * (4 * KTP_H * 32);
    const v4f* pBih1 = wsq + QIH1 + w * (4 * KTP_H * 32);
    const v4f* pBhh1 = wsq + QHH1 + w * (4 * KTP_H * 32);

    // bias for this lane's 4 columns (scalar regs; added at gate-store time)
    float bias0r[4], bias1r[4];
#pragma unroll
    for (int nt = 0; nt < 4; ++nt) {
        int col = (w << 6) + (nt << 4) + (lane & 15);
        bias0r[nt] = ws[OFF_B0 + col];
        bias1r[nt] = ws[OFF_B1 + col];
    }

    // elementwise ownership: column jown, rows mb..mb+7
    const int jown  = tid & 127;
    const int mb    = (tid >> 7) << 3;
    const int ktpj  = jown >> 3;
    const int lanej = ((jown >> 1) & 1) << 4;
    const int slotj = (jown & 1) + (((jown >> 2) & 1) << 1);
    float c0r[8], c1r[8];

    // ---- init states from h0/c0 inputs (fragment layout for h) ----
    {
        float* h0f = (float*)sm_h0;
        float* h1f = (float*)sm_h1;
#pragma unroll
        for (int i = 0; i < 8; ++i) {
            int e = tid + (i << 8);      // 2048 = 16*128
            int m = e >> 7, j = e & 127;
            int fo = (j >> 3) * 128 + (((((j >> 1) & 1) << 4)) + m) * 4 +
                     ((j & 1) + (((j >> 2) & 1) << 1));
            h0f[fo] = h0in[(b0 + m) * H + j];
            h1f[fo] = h0in[B * H + (b0 + m) * H + j];
            c0r[i]  = c0in[(b0 + mb + i) * H + jown];
            c1r[i]  = c0in[B * H + (b0 + mb + i) * H + jown];
        }
    }

    // ---- precompute x staging offsets (constant across t) ----
    int xsrc[10], xdst[10];
#pragma unroll
    for (int i = 0; i < 10; ++i) {
        int e = tid + (i << 8);          // 2560 = 16*160
        int m = e / D, k = e - m * D;
        xsrc[i] = (b0 + m) * (T * D) + k;
        xdst[i] = (k >> 3) * 128 + (((((k >> 1) & 1) << 4)) + m) * 4 +
                  ((k & 1) + (((k >> 2) & 1) << 1));
    }

    __syncthreads();

    // Uniform zero laundered through asm each iteration: makes weight-load
    // addresses loop-variant (defeats LICM / register-hoisting of weights).
    int lz = 0;

    for (int t = 0; t < T; ++t) {
        asm volatile("" : "+s"(lz));
        const v4f* pB0 = pBih0 + lz;
        const v4f* pB1 = pBhh0 + lz;
        const v4f* pB2 = pBih1 + lz;
        const v4f* pB3 = pBhh1 + lz;

        // ---- stage x_t into A-fragment layout ----
        {
            const float* xt = x + t * D;
            float* xf = (float*)sm_x;
#pragma unroll
            for (int i = 0; i < 10; ++i) xf[xdst[i]] = xt[xsrc[i]];
        }
        __syncthreads();

        // ---- layer 0 gates: x_t @ Wih0^T + h0 @ Whh0^T + b0 ----
        {
            v8f acc[4] = {};   // inline-0 C operand on first WMMA of chain
            gemm_part<KTP_X>(sm_x,  pB0, lane, acc);
            gemm_part<KTP_H>(sm_h0, pB1, lane, acc);
            store_gates(acc, bias0r, sm_g, w, lane);
        }
        __syncthreads();

        // ---- layer 0 cell update (writes new h into sm_h0) ----
        cell_update(sm_g, c0r, (float*)sm_h0, jown, mb, ktpj, lanej, slotj,
                    nullptr);
        __syncthreads();

        // ---- layer 1 gates: h0_t @ Wih1^T + h1 @ Whh1^T + b1 ----
        {
            v8f acc[4] = {};
            gemm_part<KTP_H>(sm_h0, pB2, lane, acc);
            gemm_part<KTP_H>(sm_h1, pB3, lane, acc);
            store_gates(acc, bias1r, sm_g, w, lane);
        }
        __syncthreads();

        // ---- layer 1 cell update (+ final output on last step) ----
        cell_update(sm_g, c1r, (float*)sm_h1, jown, mb, ktpj, lanej, slotj,
                    (t == T - 1) ? (out + b0 * H) : nullptr);
        __syncthreads();
    }
}

// ---------------------------------------------------------------------------
extern "C" void kernel_launch(void* const* d_in, const int* in_sizes, int n_in,
                              void* d_out, int out_size, void* d_ws,
                              size_t ws_size, hipStream_t stream) {
    (void)in_sizes; (void)n_in; (void)out_size; (void)ws_size;
    const float* x    = (const float*)d_in[0];
    const float* h0   = (const float*)d_in[1];
    const float* c0   = (const float*)d_in[2];
    const float* Wih0 = (const float*)d_in[3];
    const float* Whh0 = (const float*)d_in[4];
    const float* bih0 = (const float*)d_in[5];
    const float* bhh0 = (const float*)d_in[6];
    const float* Wih1 = (const float*)d_in[7];
    const float* Whh1 = (const float*)d_in[8];
    const float* bih1 = (const float*)d_in[9];
    const float* bhh1 = (const float*)d_in[10];
    float* ws = (float*)d_ws;

    // pack weights+biases: (69632 weight quads + 1024 bias floats) threads
    lstm_pack<<<(QTOTAL + 1024 + 255) / 256, 256, 0, stream>>>(
        Wih0, Whh0, Wih1, Whh1, bih0, bhh0, bih1, bhh1, ws);

    // fused recurrence: 16 batch tiles x 256 threads (8 wave32 per WG)
    lstm_fused<<<B / 16, 256, 0, stream>>>(x, h0, c0, ws, (float*)d_out);
}